// Projection_49976239456834
// MI455X (gfx1250) — compile-verified
//
#include <hip/hip_runtime.h>
#include <stdint.h>

#define PBD_N      16384
#define PBD_NT     1024
#define PBD_ITERS  3
#define PBD_EPS    1e-7f
#define PBD_K      8
#define PBD_ROUNDS (PBD_N / (PBD_NT * PBD_K))   // 2

// ---- gfx1250 async global<->LDS helpers (ASYNCcnt-tracked DMA path) ----
// INST_OFFSET is applied to BOTH the global address
// and the LDS address, so one
// address pair + one LDS VGPR covers all three components of a node.

__device__ __forceinline__ uint32_t lds_off(const void* p) {
  // LDS aperture: low 32 bits of the pointer are the LDS byte address
  return (uint32_t)(uintptr_t)p;
}

__device__ __forceinline__ void async_g2l_b32_x3(uint32_t lds, const void* g) {
  asm volatile("global_load_async_to_lds_b32 %0, %1, off\n\t"
               "global_load_async_to_lds_b32 %0, %1, off offset:4\n\t"
               "global_load_async_to_lds_b32 %0, %1, off offset:8"
               :: "v"(lds), "v"(g) : "memory");
}

__device__ __forceinline__ void async_l2g_b32_x3(const void* g, uint32_t lds) {
  asm volatile("global_store_async_from_lds_b32 %0, %1, off\n\t"
               "global_store_async_from_lds_b32 %0, %1, off offset:4\n\t"
               "global_store_async_from_lds_b32 %0, %1, off offset:8"
               :: "v"(g), "v"(lds) : "memory");
}

__device__ __forceinline__ void wait_async0() {
#if defined(__has_builtin)
#if __has_builtin(__builtin_amdgcn_s_wait_asynccnt)
  __builtin_amdgcn_s_wait_asynccnt(0);
#else
  asm volatile("s_wait_asynccnt 0" ::: "memory");
#endif
#else
  asm volatile("s_wait_asynccnt 0" ::: "memory");
#endif
}

// One workgroup per batch: chain lives in LDS as padded float4/node (256KB of 320KB).
extern "C" __global__ void __launch_bounds__(PBD_NT)
pbd_projection_kernel(const float* __restrict__ xin, float* __restrict__ out) {
  extern __shared__ char lds_raw[];
  float4* nodes = (float4*)lds_raw;                                   // [N] 16B each
  float*  redC  = (float*)(lds_raw + (size_t)PBD_N * sizeof(float4)); // [32]
  float*  redD  = redC + 32;                                          // [32]
  float*  sS    = redD + 32;                                          // [1]
  float*  halo  = sS + 1;                                             // [6]

  const int tid = threadIdx.x;
  const float* gx   = xin + (size_t)blockIdx.x * (PBD_N * 3);
  float*       gout = out + (size_t)blockIdx.x * (PBD_N * 3);

  // ---- Stage: packed AoS (12B/node) global -> padded AoS (16B/node) LDS ----
  {
    const float* g = gx + 3 * tid;
    uint32_t l = lds_off(nodes + tid);
    for (int i = tid; i < PBD_N; i += PBD_NT) {
      async_g2l_b32_x3(l, g);                 // offsets 0/4/8 on both sides
      g += 3 * PBD_NT;
      l += (uint32_t)(sizeof(float4) * PBD_NT);
    }
  }
  wait_async0();
  __syncthreads();

  for (int it = 0; it < PBD_ITERS; ++it) {
    // ---- Phase 1: C = sum seg^2, denom = sum |grad|^2 (b128 stencil taps) ----
    float cAcc = 0.0f, dAcc = 0.0f;
    for (int j = tid; j < PBD_N; j += PBD_NT) {
      float4 cv = nodes[j];
      float segL = 0.f, dLx = 0.f, dLy = 0.f, dLz = 0.f;
      if (j > 0) {
        float4 lv = nodes[j - 1];
        dLx = cv.x - lv.x; dLy = cv.y - lv.y; dLz = cv.z - lv.z;
        segL = dLx * dLx + dLy * dLy + dLz * dLz - 1.0f;
      }
      float segR = 0.f, dRx = 0.f, dRy = 0.f, dRz = 0.f;
      if (j < PBD_N - 1) {
        float4 rv = nodes[j + 1];
        dRx = rv.x - cv.x; dRy = rv.y - cv.y; dRz = rv.z - cv.z;
        segR = dRx * dRx + dRy * dRy + dRz * dRz - 1.0f;
        cAcc += segR * segR;                    // each segment counted once
      }
      float gxv = 4.0f * (segL * dLx - segR * dRx);
      float gyv = 4.0f * (segL * dLy - segR * dRy);
      float gzv = 4.0f * (segL * dLz - segR * dRz);
      dAcc += gxv * gxv + gyv * gyv + gzv * gzv;
    }
    // wave32 tree reduce, then cross-wave via LDS
    for (int m = 16; m > 0; m >>= 1) {
      cAcc += __shfl_xor(cAcc, m, 32);
      dAcc += __shfl_xor(dAcc, m, 32);
    }
    if ((tid & 31) == 0) { redC[tid >> 5] = cAcc; redD[tid >> 5] = dAcc; }
    __syncthreads();
    if (tid == 0) {
      float C = 0.f, Dn = 0.f;
      for (int w = 0; w < PBD_NT / 32; ++w) { C += redC[w]; Dn += redD[w]; }
      sS[0] = C / (Dn + PBD_EPS);
    }
    __syncthreads();
    const float s = sS[0];

    // ---- Phase 2: K=8 strided nodes/thread staged in registers, 2 rounds ----
    for (int r = 0; r < PBD_ROUNDS; ++r) {
      float nx[PBD_K], ny[PBD_K], nz[PBD_K];
#pragma unroll
      for (int k = 0; k < PBD_K; ++k) {
        const int j = r * (PBD_NT * PBD_K) + k * PBD_NT + tid;
        float4 cv = nodes[j];
        float lx = 0.f, ly = 0.f, lz = 0.f;
        if (k == 0 && tid == 0) {
          if (r > 0) {                    // old value of previous round's last node
            const int ps = ((r - 1) & 1) * 3;
            lx = halo[ps]; ly = halo[ps + 1]; lz = halo[ps + 2];
          }
        } else {
          float4 lv = nodes[j - 1];
          lx = lv.x; ly = lv.y; lz = lv.z;
        }
        if (k == PBD_K - 1 && tid == PBD_NT - 1) {  // stash old value for next round
          const int ws = (r & 1) * 3;
          halo[ws] = cv.x; halo[ws + 1] = cv.y; halo[ws + 2] = cv.z;
        }
        float segL = 0.f, dLx = 0.f, dLy = 0.f, dLz = 0.f;
        if (j > 0) {
          dLx = cv.x - lx; dLy = cv.y - ly; dLz = cv.z - lz;
          segL = dLx * dLx + dLy * dLy + dLz * dLz - 1.0f;
        }
        float segR = 0.f, dRx = 0.f, dRy = 0.f, dRz = 0.f;
        if (j < PBD_N - 1) {
          float4 rv = nodes[j + 1];
          dRx = rv.x - cv.x; dRy = rv.y - cv.y; dRz = rv.z - cv.z;
          segR = dRx * dRx + dRy * dRy + dRz * dRz - 1.0f;
        }
        float gxv = 4.0f * (segL * dLx - segR * dRx);
        float gyv = 4.0f * (segL * dLy - segR * dRy);
        float gzv = 4.0f * (segL * dLz - segR * dRz);
        nx[k] = cv.x - s * gxv;
        ny[k] = cv.y - s * gyv;
        nz[k] = cv.z - s * gzv;
      }
      __syncthreads();                    // all old-value reads complete
#pragma unroll
      for (int k = 0; k < PBD_K; ++k) {
        const int j = r * (PBD_NT * PBD_K) + k * PBD_NT + tid;
        nodes[j] = make_float4(nx[k], ny[k], nz[k], 0.0f);   // single b128 store
      }
      __syncthreads();                    // writes visible before next reads
    }
  }

  // ---- Drain: padded AoS LDS -> packed AoS global ----
  {
    float* g = gout + 3 * tid;
    uint32_t l = lds_off(nodes + tid);
    for (int i = tid; i < PBD_N; i += PBD_NT) {
      async_l2g_b32_x3(g, l);                 // offsets 0/4/8 on both sides
      g += 3 * PBD_NT;
      l += (uint32_t)(sizeof(float4) * PBD_NT);
    }
  }
  wait_async0();
}

extern "C" void kernel_launch(void* const* d_in, const int* in_sizes, int n_in,
                              void* d_out, int out_size, void* d_ws, size_t ws_size,
                              hipStream_t stream) {
  const float* x = (const float*)d_in[0];
  float* out = (float*)d_out;
  const int B = in_sizes[0] / (PBD_N * 3);       // 512
  const size_t shmem = (size_t)PBD_N * sizeof(float4) + 512;  // 256KB + scratch
  hipLaunchKernelGGL(pbd_projection_kernel, dim3(B), dim3(PBD_NT), shmem, stream,
                     x, out);
}